// GraphAttentionLayer_27590869909990
// MI455X (gfx1250) — compile-verified
//
#include <hip/hip_runtime.h>

typedef __attribute__((ext_vector_type(2))) float v2f;
typedef __attribute__((ext_vector_type(8))) float v8f;

#define BATCH 8
#define NN    2048
#define FIN   128
#define FOUT  64
#define ALPHA 0.2f
#define NEGINF (-1000000000.0f)

// ---------------------------------------------------------------------------
// Phase 1: Wh = h @ W^T  (per 16-row tile, fp32 WMMA 16x16x4, K-loop over 128)
//          s1 = Wh @ a[:64], s2 = Wh @ a[64:]  (butterfly reduction, wave32)
// One wave handles one 16-row tile. Grid = B*N/16 tiles, 4 waves per block.
// ---------------------------------------------------------------------------
__global__ __launch_bounds__(128) void gat_phase1(
    const float* __restrict__ h, const float* __restrict__ W,
    const float* __restrict__ a, float* __restrict__ Wh,
    float* __restrict__ s1, float* __restrict__ s2)
{
    const int wave = threadIdx.x >> 5;
    const int lane = threadIdx.x & 31;
    const int r    = lane & 15;        // row (A) / col (B,C) within tile
    const int hi   = lane >> 4;        // lane half selects K pair / row half

    const int tile = blockIdx.x * 4 + wave;   // 0 .. B*N/16-1 (exact)
    const int b    = tile >> 7;               // N/16 == 128
    const int i0   = (tile & 127) << 4;

    const float* hbase = h + ((size_t)b * NN + i0) * FIN;

    v8f acc[4] = {};   // 4 N-tiles of 16 -> 64 output features

    for (int k0 = 0; k0 < FIN; k0 += 4) {
        // A (16x4 fp32): lane r holds K = k0 + 2*hi + {0,1}  -> contiguous b64
        const float* ap = hbase + (size_t)r * FIN + k0 + 2 * hi;
        v2f av; av.x = ap[0]; av.y = ap[1];
#pragma unroll
        for (int nt = 0; nt < 4; ++nt) {
            // B (4x16): B[k][f] = W[f][k]; f = nt*16 + r, k contiguous
            const float* bp = W + (size_t)(nt * 16 + r) * FIN + k0 + 2 * hi;
            v2f bv; bv.x = bp[0]; bv.y = bp[1];
            acc[nt] = __builtin_amdgcn_wmma_f32_16x16x4_f32(
                false, av, false, bv, (short)0, acc[nt], false, false);
        }
    }

    // Store Wh tile: C layout -> lane L reg v holds [row v+8*hi][col nt*16+r]
    float* wtile = Wh + ((size_t)b * NN + i0) * FOUT;
#pragma unroll
    for (int nt = 0; nt < 4; ++nt)
#pragma unroll
        for (int v = 0; v < 8; ++v)
            wtile[(size_t)(v + 8 * hi) * FOUT + nt * 16 + r] = acc[nt][v];

    // s1/s2: per row M = v+8*hi, reduce over features (lanes within half)
#pragma unroll
    for (int v = 0; v < 8; ++v) {
        float p1 = 0.f, p2 = 0.f;
#pragma unroll
        for (int nt = 0; nt < 4; ++nt) {
            p1 += acc[nt][v] * a[nt * 16 + r];
            p2 += acc[nt][v] * a[FOUT + nt * 16 + r];
        }
#pragma unroll
        for (int off = 1; off < 16; off <<= 1) {
            p1 += __shfl_xor(p1, off, 32);
            p2 += __shfl_xor(p2, off, 32);
        }
        if (r == 0) {
            s1[(size_t)b * NN + i0 + v + 8 * hi] = p1;
            s2[(size_t)b * NN + i0 + v + 8 * hi] = p2;
        }
    }
}

// ---------------------------------------------------------------------------
// Phase 2: fused mask + online softmax + attention @ Wh.
// Block = 4 waves = one 16-row i-tile. Each wave streams a contiguous 512-col
// slice of adj (read exactly once), holding probabilities directly in the
// WMMA A-register layout. Partials merged via LDS (flash-decoding combine).
// ---------------------------------------------------------------------------
__global__ __launch_bounds__(128) void gat_phase2(
    const float* __restrict__ Wh, const float* __restrict__ s1,
    const float* __restrict__ s2, const int* __restrict__ adj,
    float* __restrict__ out)
{
    __shared__ float lm[4][16];
    __shared__ float ll[4][16];
    __shared__ float lacc[4][16][FOUT];

    const int wave = threadIdx.x >> 5;
    const int lane = threadIdx.x & 31;
    const int r    = lane & 15;
    const int hi   = lane >> 4;

    const int b  = blockIdx.x >> 7;          // N/16 == 128
    const int i0 = (blockIdx.x & 127) << 4;

    const int*   adjrow = adj + ((size_t)b * NN + (i0 + r)) * NN;
    const float* Whb    = Wh + (size_t)b * NN * FOUT;
    const float* s2b    = s2 + (size_t)b * NN;
    const float  s1r    = s1[(size_t)b * NN + i0 + r];

    float m = NEGINF;   // running row max (replicated in both lane halves)
    float l = 0.f;      // running row denom
    v8f acc[4] = {};    // running weighted sum (C layout, 16x64)

    const int c0 = wave * 32;                // contiguous 32 chunks per wave
    for (int c = c0; c < c0 + 32; ++c) {
        const int j0 = c << 4;
        float ev[4][2];
        float cmax = NEGINF;
#pragma unroll
        for (int kk = 0; kk < 4; ++kk) {
            const int cb = j0 + kk * 4 + 2 * hi;   // A-layout columns
            const int2   aij = *(const int2*)(adjrow + cb);
            const float2 sv  = *(const float2*)(s2b + cb);
            const float x0 = s1r + sv.x;
            const float x1 = s1r + sv.y;
            float e0 = fmaxf(x0, ALPHA * x0);      // leaky_relu (alpha < 1)
            float e1 = fmaxf(x1, ALPHA * x1);
            e0 = aij.x ? e0 : NEGINF;
            e1 = aij.y ? e1 : NEGINF;
            ev[kk][0] = e0; ev[kk][1] = e1;
            cmax = fmaxf(cmax, fmaxf(e0, e1));
        }
        cmax = fmaxf(cmax, __shfl_xor(cmax, 16, 32));   // row max across halves
        const float mnew  = fmaxf(m, cmax);
        const float scale = __expf(m - mnew);

        float psum = 0.f;
        v2f pa[4];
#pragma unroll
        for (int kk = 0; kk < 4; ++kk) {
            const float p0 = __expf(ev[kk][0] - mnew);
            const float p1 = __expf(ev[kk][1] - mnew);
            psum += p0 + p1;
            pa[kk].x = p0; pa[kk].y = p1;
        }
        psum += __shfl_xor(psum, 16, 32);
        l = l * scale + psum;
        m = mnew;

        // Rescale accumulator rows: broadcast scale[row v+8*hi] from lane v+8*hi
#pragma unroll
        for (int v = 0; v < 8; ++v) {
            const float sc = __shfl(scale, v + 8 * hi, 32);
#pragma unroll
            for (int nt = 0; nt < 4; ++nt) acc[nt][v] *= sc;
        }

        // acc += P(16x16) @ Wh_chunk(16x64): 4 K-steps x 4 N-tiles of fp32 WMMA
#pragma unroll
        for (int kk = 0; kk < 4; ++kk) {
            const float* brow = Whb + (size_t)(j0 + kk * 4 + 2 * hi) * FOUT;
#pragma unroll
            for (int nt = 0; nt < 4; ++nt) {
                v2f bv;
                bv.x = brow[nt * 16 + r];
                bv.y = brow[FOUT + nt * 16 + r];
                acc[nt] = __builtin_amdgcn_wmma_f32_16x16x4_f32(
                    false, pa[kk], false, bv, (short)0, acc[nt], false, false);
            }
        }
    }

    // Publish per-wave partials
    if (hi == 0) { lm[wave][r] = m; ll[wave][r] = l; }
#pragma unroll
    for (int nt = 0; nt < 4; ++nt)
#pragma unroll
        for (int v = 0; v < 8; ++v)
            lacc[wave][v + 8 * hi][nt * 16 + r] = acc[nt][v];
    __syncthreads();

    // Combine 4 waves and write output (coalesced)
    float* otile = out + ((size_t)b * NN + i0) * FOUT;
    for (int idx = threadIdx.x; idx < 16 * FOUT; idx += 128) {
        const int rr = idx >> 6;
        const int ff = idx & 63;
        const float mm = fmaxf(fmaxf(lm[0][rr], lm[1][rr]),
                               fmaxf(lm[2][rr], lm[3][rr]));
        float lsum = 0.f, asum = 0.f;
#pragma unroll
        for (int w = 0; w < 4; ++w) {
            const float sc = __expf(lm[w][rr] - mm);
            lsum += ll[w][rr] * sc;
            asum += lacc[w][rr][ff] * sc;
        }
        otile[(size_t)rr * FOUT + ff] = asum / lsum;
    }
}

// ---------------------------------------------------------------------------
extern "C" void kernel_launch(void* const* d_in, const int* in_sizes, int n_in,
                              void* d_out, int out_size, void* d_ws, size_t ws_size,
                              hipStream_t stream)
{
    (void)in_sizes; (void)n_in; (void)out_size; (void)ws_size;
    const float* h   = (const float*)d_in[0];
    const int*   adj = (const int*)d_in[1];
    const float* W   = (const float*)d_in[2];
    const float* a   = (const float*)d_in[3];
    float* out = (float*)d_out;

    float* Wh = (float*)d_ws;                                 // B*N*FOUT floats
    float* s1 = Wh + (size_t)BATCH * NN * FOUT;               // B*N floats
    float* s2 = s1 + (size_t)BATCH * NN;                      // B*N floats

    gat_phase1<<<(BATCH * NN / 16) / 4, 128, 0, stream>>>(h, W, a, Wh, s1, s2);
    gat_phase2<<<BATCH * NN / 16, 128, 0, stream>>>(Wh, s1, s2, adj, out);
}